// DocMixin_56444460204499
// MI455X (gfx1250) — compile-verified
//
#include <hip/hip_runtime.h>

// CDNA5 / gfx1250, wave32. Bandwidth-bound problem: ~550MB HBM traffic vs
// 0.27 GFLOP -> stay in f32 end-to-end and use V_WMMA_F32_16X16X4_F32 for the
// segment-weighted GEMM (full precision, matrix pipe does the FMAs).
// All pair loads are native-vector <2 x i32> so they lower to a single
// unconditional global_load_b64 (no load sinking into select arms).

typedef float v2f __attribute__((ext_vector_type(2)));
typedef float v4f __attribute__((ext_vector_type(4)));
typedef float v8f __attribute__((ext_vector_type(8)));
typedef int   v2i __attribute__((ext_vector_type(2)));

// ---------------------------------------------------------------------------
// K1: scores[i] = dot(seq_feats[i,:], W_attn) + b   (one wave per row)
// feats is a 256MB single-pass stream -> non-temporal loads so it doesn't
// evict seq_logits working set from L2 (192MB) before K4 runs.
// ---------------------------------------------------------------------------
__global__ __launch_bounds__(256) void k1_scores(
        const float* __restrict__ feats,   // [N,H]
        const float* __restrict__ Wv,      // [H]
        const float* __restrict__ bv,      // [1]
        float* __restrict__ scores,        // [N]
        int N, int H) {
    const int wave = (blockIdx.x * blockDim.x + threadIdx.x) >> 5;
    const int lane = threadIdx.x & 31;
    if (wave >= N) return;
    const v4f* __restrict__ f4 = reinterpret_cast<const v4f*>(feats + (size_t)wave * H);
    const v4f* __restrict__ w4 = reinterpret_cast<const v4f*>(Wv);
    const int nv = H >> 2;
    float s = 0.f;
    for (int j = lane; j < nv; j += 32) {          // 512B/wave/iter, b128 loads
        v4f a = __builtin_nontemporal_load(&f4[j]);
        v4f w = w4[j];                             // L2-resident broadcast
        s += a.x * w.x + a.y * w.y + a.z * w.z + a.w * w.w;
    }
    #pragma unroll
    for (int off = 16; off; off >>= 1) s += __shfl_xor(s, off, 32);
    if (lane == 0) scores[wave] = s + bv[0];
}

// ---------------------------------------------------------------------------
// K2: per-doc max / sum-exp over the sorted segment run + doc start offsets
// ---------------------------------------------------------------------------
__device__ __forceinline__ int lower_bound_i32(const int* __restrict__ a, int n, int key) {
    int lo = 0, hi = n;
    while (lo < hi) {
        int mid = (lo + hi) >> 1;
        if (a[mid] < key) lo = mid + 1; else hi = mid;
    }
    return lo;
}

__global__ __launch_bounds__(256) void k2_segstats(
        const int*   __restrict__ seg,     // [N] sorted
        const float* __restrict__ scores,  // [N]
        float* __restrict__ mx,            // [D]
        float* __restrict__ denom,         // [D]
        int*   __restrict__ doc_lo,        // [D]
        int N, int D) {
    const int d = blockIdx.x * blockDim.x + threadIdx.x;
    if (d >= D) return;
    const int lo = lower_bound_i32(seg, N, d);
    const int hi = lower_bound_i32(seg, N, d + 1);
    doc_lo[d] = lo;
    float m = -3.402823466e38f;
    for (int i = lo; i < hi; ++i) m = fmaxf(m, scores[i]);
    float s = 0.f;
    for (int i = lo; i < hi; ++i) s += __expf(scores[i] - m);
    mx[d] = m;
    denom[d] = s;
}

// ---------------------------------------------------------------------------
// K3: per-sentence softmax weight, packed with its segment id so K4's A-side
// data is a single b64 load per lane per K-step.
// ---------------------------------------------------------------------------
__global__ __launch_bounds__(256) void k3_weights(
        const int*   __restrict__ seg,
        const float* __restrict__ scores,
        const float* __restrict__ mx,
        const float* __restrict__ denom,
        v2i* __restrict__ pairs,           // [N] {seg_id, bits(w)}
        int N) {
    const int i = blockIdx.x * blockDim.x + threadIdx.x;
    if (i >= N) return;
    const int d = seg[i];
    const float wv = __expf(scores[i] - mx[d]) / denom[d];
    v2i pr;
    pr.x = d;
    pr.y = __float_as_int(wv);
    pairs[i] = pr;                         // single b64 store
}

// ---------------------------------------------------------------------------
// K4: doc_logits[16-doc group] += A(16x4 weights) x B(4x16 logits) via
//     v_wmma_f32_16x16x4_f32.  Block = 8 waves = one doc group; waves stripe
//     over the class tiles, so each sentence row streams from HBM once and
//     re-touches hit WGP$/L2.
//
//  A (32b 16x4): lanes 0-15 -> M=lane, V0=K0,V1=K1; lanes 16-31 -> M=lane-16,
//  V0=K2,V1=K3.  B (4x16): lanes 0-15 -> N=lane, K0/K1; lanes 16-31 ->
//  N=lane-16, K2/K3.  Per lane kbase=(lane>=16)*2; both A and B use sentence
//  indices s+kbase, s+kbase+1.  C/D: VGPR r -> M=r+8*(lane>=16), N=lane&15.
// ---------------------------------------------------------------------------
__global__ __launch_bounds__(256, 2) void k4_docgemm(
        const float* __restrict__ logits,  // [N,C]
        const v2i*   __restrict__ pairs,   // [N] {seg, bits(w)}
        const int*   __restrict__ doc_lo,  // [D]
        const float* __restrict__ mask,    // [C]
        float* __restrict__ out,           // [D,C]
        int N, int C, int D) {
    const int g    = blockIdx.x;           // doc group id
    const int base = g * 16;
    const int lane = threadIdx.x & 31;
    const int wid  = threadIdx.x >> 5;
    const int m15  = lane & 15;
    const int hi   = lane >> 4;            // 0 -> K0/K1 half, 1 -> K2/K3 half
    const int kb   = hi * 2;
    const int docm = base + m15;           // doc this lane matches in A

    const int start = doc_lo[base];
    const int nxt   = base + 16;
    const int end   = (nxt < D) ? doc_lo[nxt] : N;   // sorted -> contiguous run
    const int NT    = (C + 15) >> 4;
    const size_t stepL = (size_t)4 * C;

    for (int t = wid; t < NT; t += 8) {
        const int  c  = t * 16 + m15;
        const bool cv = (c < C);
        const int  cc = cv ? c : (C - 1);  // clamped column (garbage cols never stored)
        v8f acc = {};

        // per-lane streaming pointers (constant-stride -> strength-reduced)
        const v2i*   pp = pairs  + (start + kb);
        const float* lp = logits + (size_t)(start + kb) * C + cc;

        // ---- hot loop: all 4 sentence slots in-range, no predication ----
        int s = start;
        #pragma unroll 2
        for (; s + 4 <= end; s += 4) {
            v2i   p0 = pp[0];              // one unconditional global_load_b64
            v2i   p1 = pp[1];
            float b0 = lp[0];
            float b1 = lp[(size_t)C];
            v2f av, bm;
            av.x = (p0.x == docm) ? __int_as_float(p0.y) : 0.f;   // v_cndmask
            av.y = (p1.x == docm) ? __int_as_float(p1.y) : 0.f;
            bm.x = b0;
            bm.y = b1;
            acc = __builtin_amdgcn_wmma_f32_16x16x4_f32(
                false, av, false, bm, (short)0, acc, false, false);
            pp += 4;
            lp += stepL;
        }
        // ---- tail (<4 sentences): clamp addresses, zero A by select ----
        if (s < end) {
            const int  i0 = s + kb;
            const int  i1 = i0 + 1;
            const bool v0 = (i0 < end);
            const bool v1 = (i1 < end);
            const int  ic0 = v0 ? i0 : start;   // safe: run non-empty here
            const int  ic1 = v1 ? i1 : start;
            v2i   p0 = pairs[ic0];
            v2i   p1 = pairs[ic1];
            float b0 = logits[(size_t)ic0 * C + cc];
            float b1 = logits[(size_t)ic1 * C + cc];
            v2f av, bm;
            av.x = (v0 && p0.x == docm) ? __int_as_float(p0.y) : 0.f;
            av.y = (v1 && p1.x == docm) ? __int_as_float(p1.y) : 0.f;
            bm.x = b0;   // garbage rows contribute 0: their A column is all-zero
            bm.y = b1;
            acc = __builtin_amdgcn_wmma_f32_16x16x4_f32(
                false, av, false, bm, (short)0, acc, false, false);
        }

        if (cv) {
            const float off = (mask[c] - 1.0f) * 1e10f;
            #pragma unroll
            for (int r = 0; r < 8; ++r) {
                const int doc = base + r + hi * 8;
                if (doc < D)
                    __builtin_nontemporal_store(acc[r] + off, &out[(size_t)doc * C + c]);
            }
        }
    }
}

// ---------------------------------------------------------------------------
extern "C" void kernel_launch(void* const* d_in, const int* in_sizes, int n_in,
                              void* d_out, int out_size, void* d_ws, size_t ws_size,
                              hipStream_t stream) {
    const float* feats  = (const float*)d_in[0];   // [N,H]
    const float* logits = (const float*)d_in[1];   // [N,C]
    const float* Wv     = (const float*)d_in[2];   // [H,1]
    const float* bv     = (const float*)d_in[3];   // [1]
    const float* mask   = (const float*)d_in[4];   // [C]
    const int*   seg    = (const int*)d_in[5];     // [N]
    // d_in[6] = num_docs (device scalar) -- D derived from out_size instead.

    const int C = in_sizes[4];
    const int N = in_sizes[1] / C;
    const int H = in_sizes[0] / N;
    const int D = out_size / C;

    // workspace layout (8B-aligned first): pairs[N] v2i | scores[N] | mx[D]
    //                                      | denom[D] | doc_lo[D]
    v2i*   pairs  = (v2i*)d_ws;
    float* scores = (float*)(pairs + N);
    float* mx     = scores + N;
    float* denom  = mx + D;
    int*   doc_lo = (int*)(denom + D);

    // K1: one wave per row, 8 waves per block
    {
        const int rows_per_block = 8;
        const int blocks = (N + rows_per_block - 1) / rows_per_block;
        k1_scores<<<blocks, 256, 0, stream>>>(feats, Wv, bv, scores, N, H);
    }
    // K2: one thread per doc
    k2_segstats<<<(D + 255) / 256, 256, 0, stream>>>(seg, scores, mx, denom, doc_lo, N, D);
    // K3: one thread per sentence (packs {seg, w})
    k3_weights<<<(N + 255) / 256, 256, 0, stream>>>(seg, scores, mx, denom, pairs, N);
    // K4: one block per 16-doc group
    {
        const int groups = (D + 15) / 16;
        k4_docgemm<<<groups, 256, 0, stream>>>(logits, pairs, doc_lo, mask,
                                               (float*)d_out, N, C, D);
    }
}